// SelfMixLayer_4045859192944
// MI455X (gfx1250) — compile-verified
//
#include <hip/hip_runtime.h>
#include <hip/hip_bf16.h>
#include <math.h>

typedef __attribute__((ext_vector_type(2))) float v2f;
typedef __attribute__((ext_vector_type(8))) float v8f;

// ---------------------------------------------------------------------------
// Degrees (0,1,2,3): dims (1,3,5,7), offsets (0,1,4,9), M_TOT = 16.
// TRIL pairs (l2,l1): (1,0) (2,0) (2,1) (3,0) (3,1) (3,2)  -> P = 6
// K = dim2*dim1 padded to x4 -> S k-steps; T0 = cumulative step base.
// ---------------------------------------------------------------------------
constexpr int P_OFF2[6] = {1, 4, 4, 9, 9, 9};
constexpr int P_OFF1[6] = {0, 0, 1, 0, 1, 4};
constexpr int P_D1[6]   = {1, 1, 3, 1, 3, 5};
constexpr int P_K[6]    = {3, 5, 15, 7, 21, 35};
constexpr int P_S[6]    = {1, 2, 4, 2, 6, 9};     // total 24 wmma k-steps
constexpr int P_T0[6]   = {0, 1, 3, 7, 9, 15};

// l3 of global index p (p=0 ->0; 1..3 ->1; 4..8 ->2; 9..15 ->3)
constexpr int L3LO[8] = {0, 1, 1, 1, 2, 2, 2, 2}; // p = i      (lanes 0-15)
constexpr int L3HI[8] = {2, 3, 3, 3, 3, 3, 3, 3}; // p = i + 8  (lanes 16-31)

// ---------------------------------------------------------------------------
// Init kernel 1: dense real-CG tensor cg[p3][p2][p1] (16x16x16 f32) in d_ws.
// ---------------------------------------------------------------------------
__device__ inline double dfact(int v) {
    double r = 1.0;
    for (int i = 2; i <= v; ++i) r *= (double)i;
    return r;
}

__device__ double su2cg(int j1, int j2, int j3, int m1, int m2, int m3) {
    if (m3 != m1 + m2) return 0.0;
    int lo = (j1 > j2) ? (j1 - j2) : (j2 - j1);
    if (j3 < lo || j3 > j1 + j2) return 0.0;
    double pref = sqrt((2.0 * j3 + 1.0) * dfact(j3 + j1 - j2) * dfact(j3 - j1 + j2) *
                       dfact(j1 + j2 - j3) / dfact(j1 + j2 + j3 + 1) *
                       dfact(j3 + m3) * dfact(j3 - m3) * dfact(j1 - m1) *
                       dfact(j1 + m1) * dfact(j2 - m2) * dfact(j2 + m2));
    double s = 0.0;
    for (int k = 0; k <= j1 + j2 - j3; ++k) {
        if (j1 - m1 - k < 0 || j2 + m2 - k < 0 || j3 - j2 + m1 + k < 0 ||
            j3 - j1 - m2 + k < 0)
            continue;
        double t = 1.0 / (dfact(k) * dfact(j1 + j2 - j3 - k) * dfact(j1 - m1 - k) *
                          dfact(j2 + m2 - k) * dfact(j3 - j2 + m1 + k) *
                          dfact(j3 - j1 - m2 + k));
        s += (k & 1) ? -t : t;
    }
    return pref * s;
}

__device__ void qent(int l, int row, int col, double& outr, double& outi) {
    const double inv = 0.70710678118654752440;
    double re = 0.0, im = 0.0;
    int mm = row - l;
    if (mm < 0) {
        int am = -mm;
        if (col == l + am) re = inv;
        else if (col == l - am) im = -inv;
    } else if (mm == 0) {
        if (col == l) re = 1.0;
    } else {
        double sg = (mm & 1) ? -1.0 : 1.0;
        if (col == l + mm) re = sg * inv;
        else if (col == l - mm) im = sg * inv;
    }
    double fr, fi;
    switch (l & 3) {  // (-i)^l
        case 0: fr = 1.0;  fi = 0.0;  break;
        case 1: fr = 0.0;  fi = -1.0; break;
        case 2: fr = -1.0; fi = 0.0;  break;
        default: fr = 0.0; fi = 1.0;  break;
    }
    outr = re * fr - im * fi;
    outi = re * fi + im * fr;
}

__device__ inline void pdecomp(int p, int& l, int& idx) {
    if (p == 0)      { l = 0; idx = p;     }
    else if (p < 4)  { l = 1; idx = p - 1; }
    else if (p < 9)  { l = 2; idx = p - 4; }
    else             { l = 3; idx = p - 9; }
}

__global__ void cg_init_kernel(float* __restrict__ cg) {
    int tid = blockIdx.x * blockDim.x + threadIdx.x;
    if (tid >= 4096) return;
    int p3 = tid >> 8, p2 = (tid >> 4) & 15, p1 = tid & 15;
    int l3, m, l2, li, l1, j;
    pdecomp(p3, l3, m);
    pdecomp(p2, l2, li);
    pdecomp(p1, l1, j);
    double acc = 0.0;
    for (int i = 0; i <= 2 * l1; ++i) {
        for (int k = 0; k <= 2 * l2; ++k) {
            int m1 = i - l1, m2 = k - l2;
            int nn = m1 + m2 + l3;
            if (nn < 0 || nn > 2 * l3) continue;
            double C = su2cg(l1, l2, l3, m1, m2, m1 + m2);
            if (C == 0.0) continue;
            double q1r, q1i, q2r, q2i, q3r, q3i;
            qent(l1, i, j, q1r, q1i);
            qent(l2, k, li, q2r, q2i);
            qent(l3, nn, m, q3r, q3i);
            double tr = q1r * q2r - q1i * q2i;
            double ti = q1r * q2i + q1i * q2r;
            acc += C * (tr * q3r + ti * q3i);
        }
    }
    cg[tid] = (float)acc;
}

// ---------------------------------------------------------------------------
// Init kernel 2: pre-swizzle A operands into WMMA lane layout.
// aw[t][lane][j]  (t = 0..23 k-steps, j = 0..1 A-VGPRs):
//   lane = half*16 + m ; element = CG[m][off2 + k/d1][off1 + k%d1],
//   k = 4*s + 2*half + j, zero when k >= K (padding).
// Main kernel loads av[t] as one coalesced b64 per step: zero address math.
// ---------------------------------------------------------------------------
__global__ void aw_init_kernel(const float* __restrict__ cg, float* __restrict__ aw) {
    int tid = blockIdx.x * blockDim.x + threadIdx.x;   // 24*32 = 768
    if (tid >= 768) return;
    const int t = tid >> 5, lane = tid & 31;
    const int m = lane & 15, half = lane >> 4;
    const int t0[6]   = {0, 1, 3, 7, 9, 15};
    const int kk[6]   = {3, 5, 15, 7, 21, 35};
    const int d1[6]   = {1, 1, 3, 1, 3, 5};
    const int of2[6]  = {1, 4, 4, 9, 9, 9};
    const int of1[6]  = {0, 0, 1, 0, 1, 4};
    int pr = 5;
    for (int q = 0; q < 5; ++q)
        if (t < t0[q + 1]) { pr = q; break; }
    const int s = t - t0[pr];
    for (int j = 0; j < 2; ++j) {
        const int k = 4 * s + 2 * half + j;
        float v = 0.0f;
        if (k < kk[pr])
            v = cg[m * 256 + (of2[pr] + k / d1[pr]) * 16 + (of1[pr] + k % d1[pr])];
        aw[tid * 2 + j] = v;
    }
}

// ---------------------------------------------------------------------------
// Per-pair GEMM micro-step:  D(16x16) += A_pair(16x4) x Y(4x16) over k-steps.
// B rows: lanes 0-15 carry K=4s+{0,1}, lanes 16-31 K=4s+{2,3}; N = lane&15.
// All x-register indices are compile-time after unroll.
// ---------------------------------------------------------------------------
template <int PR>
__device__ __forceinline__ v8f pair_wmma(const v2f (&av)[24], const float (&xv)[16],
                                         bool hi) {
    constexpr int OFF2 = P_OFF2[PR], OFF1 = P_OFF1[PR], D1 = P_D1[PR];
    constexpr int K = P_K[PR], S = P_S[PR], T0 = P_T0[PR];
    v8f d = {0, 0, 0, 0, 0, 0, 0, 0};
#pragma unroll
    for (int s = 0; s < S; ++s) {
        const int k0 = 4 * s, k1 = 4 * s + 1, k2 = 4 * s + 2, k3 = 4 * s + 3;
        const float p0 = (k0 < K) ? xv[OFF2 + k0 / D1] * xv[OFF1 + k0 % D1] : 0.0f;
        const float p1 = (k1 < K) ? xv[OFF2 + k1 / D1] * xv[OFF1 + k1 % D1] : 0.0f;
        const float p2 = (k2 < K) ? xv[OFF2 + k2 / D1] * xv[OFF1 + k2 % D1] : 0.0f;
        const float p3 = (k3 < K) ? xv[OFF2 + k3 / D1] * xv[OFF1 + k3 % D1] : 0.0f;
        v2f b;
        b.x = hi ? p2 : p0;
        b.y = hi ? p3 : p1;
        d = __builtin_amdgcn_wmma_f32_16x16x4_f32(false, av[T0 + s], false, b,
                                                  (short)0, d, false, false);
    }
    return d;
}

// ---------------------------------------------------------------------------
// Main kernel: one wave owns a fixed 16-wide f-tile and a run of atoms.
// FC > 0: compile-time F (immediate load/store offsets). FC == 0: runtime F.
// ---------------------------------------------------------------------------
template <int FC>
__global__ __launch_bounds__(256) void selfmix_kernel(
    const float* __restrict__ x, const float* __restrict__ s_l3,
    const float* __restrict__ k_l3, const float* __restrict__ aw,
    float* __restrict__ out, int n_atoms, int Frt, int ftiles, int atomsT) {
    const int F = (FC > 0) ? FC : Frt;
    const int w = blockIdx.x * 8 + (threadIdx.x >> 5);
    const int lane = threadIdx.x & 31;
    const int fi = w % ftiles;
    const int chunk = w / ftiles;
    const int a0 = chunk * atomsT;
    if (a0 >= n_atoms) return;                       // whole-wave exit
    const int aend = (a0 + atomsT < n_atoms) ? (a0 + atomsT) : n_atoms;
    const bool hi = lane >= 16;
    const int col = lane & 15;
    const int f = (fi << 4) + col;

    // --- A operands: 24 coalesced b64 loads, issued up front -------------
    v2f av[24];
    {
        const float* ap = aw + lane * 2;
#pragma unroll
        for (int t = 0; t < 24; ++t)
            av[t] = *(const v2f*)(ap + t * 64);
    }

    // --- s/k coefficients: constant over the atom loop (f fixed) ---------
    float sreg[8][6], kreg[8];
#pragma unroll
    for (int i = 0; i < 8; ++i) {
        const int l3 = hi ? L3HI[i] : L3LO[i];
        kreg[i] = k_l3[l3 * F + f];
#pragma unroll
        for (int pr = 0; pr < 6; ++pr)
            sreg[i][pr] = s_l3[(l3 * 6 + pr) * F + f];
    }

    // --- software-pipelined atom loop ------------------------------------
    float xv[16];
    {
        const float* xb = x + (size_t)a0 * 16 * F + f;
#pragma unroll
        for (int p = 0; p < 16; ++p) xv[p] = xb[p * F];
    }

    for (int na = a0; na < aend; ++na) {
        float xn[16];
        if (na + 1 < aend) {                         // prefetch next atom
            const float* xb = x + (size_t)(na + 1) * 16 * F + f;
#pragma unroll
            for (int p = 0; p < 16; ++p) xn[p] = xb[p * F];
        }

        float o[8];
#pragma unroll
        for (int i = 0; i < 8; ++i)
            o[i] = kreg[i] * (hi ? xv[i + 8] : xv[i]);

        {
            v8f d = pair_wmma<0>(av, xv, hi);
#pragma unroll
            for (int i = 0; i < 8; ++i) o[i] = fmaf(sreg[i][0], d[i], o[i]);
        }
        {
            v8f d = pair_wmma<1>(av, xv, hi);
#pragma unroll
            for (int i = 0; i < 8; ++i) o[i] = fmaf(sreg[i][1], d[i], o[i]);
        }
        {
            v8f d = pair_wmma<2>(av, xv, hi);
#pragma unroll
            for (int i = 0; i < 8; ++i) o[i] = fmaf(sreg[i][2], d[i], o[i]);
        }
        {
            v8f d = pair_wmma<3>(av, xv, hi);
#pragma unroll
            for (int i = 0; i < 8; ++i) o[i] = fmaf(sreg[i][3], d[i], o[i]);
        }
        {
            v8f d = pair_wmma<4>(av, xv, hi);
#pragma unroll
            for (int i = 0; i < 8; ++i) o[i] = fmaf(sreg[i][4], d[i], o[i]);
        }
        {
            v8f d = pair_wmma<5>(av, xv, hi);
#pragma unroll
            for (int i = 0; i < 8; ++i) o[i] = fmaf(sreg[i][5], d[i], o[i]);
        }

        float* ob = out + ((size_t)na * 16 + (hi ? 8 : 0)) * F + f;
#pragma unroll
        for (int i = 0; i < 8; ++i) ob[i * F] = o[i];

#pragma unroll
        for (int p = 0; p < 16; ++p) xv[p] = xn[p];
    }
}

// ---------------------------------------------------------------------------
// Inputs (setup_inputs order): 0:x 1:rbf_ij 2:sph_ij 3:phi_r_cut 4:idx_i
// 5:idx_j 6:pair_mask 7:s_l3 8:k_l3.  Only 0,7,8 feed the output.
// ---------------------------------------------------------------------------
extern "C" void kernel_launch(void* const* d_in, const int* in_sizes, int n_in,
                              void* d_out, int out_size, void* d_ws, size_t ws_size,
                              hipStream_t stream) {
    (void)n_in; (void)out_size; (void)ws_size;
    const float* x    = (const float*)d_in[0];
    const float* s_l3 = (const float*)d_in[7];
    const float* k_l3 = (const float*)d_in[8];
    float* out = (float*)d_out;
    float* cg  = (float*)d_ws;                       // 16 KB
    float* aw  = (float*)d_ws + 4096;                // 6 KB swizzled A table

    const int F = in_sizes[8] / 4;                   // k_l3 is (4, F)
    const int n = in_sizes[0] / (16 * F);            // x is (n, 16, F)
    const int ftiles = F / 16;
    const int atomsT = 8;                            // atoms per wave
    const int nchunks = (n + atomsT - 1) / atomsT;
    const int nwaves = ftiles * nchunks;
    const int blocks = (nwaves + 7) / 8;             // 8 waves / block

    hipLaunchKernelGGL(cg_init_kernel, dim3(16), dim3(256), 0, stream, cg);
    hipLaunchKernelGGL(aw_init_kernel, dim3(3), dim3(256), 0, stream, cg, aw);

    if (F == 128) {
        hipLaunchKernelGGL((selfmix_kernel<128>), dim3(blocks), dim3(256), 0, stream,
                           x, s_l3, k_l3, aw, out, n, F, ftiles, atomsT);
    } else {
        hipLaunchKernelGGL((selfmix_kernel<0>), dim3(blocks), dim3(256), 0, stream,
                           x, s_l3, k_l3, aw, out, n, F, ftiles, atomsT);
    }
}